// DistributedVQE_nparts_12859132084966
// MI455X (gfx1250) — compile-verified
//
#include <hip/hip_runtime.h>
#include <hip/hip_bf16.h>

// Distributed VQE: 2 parts x 20 qubits, depth-2 HEA + 6 Z-term expvals.
// Strategy (MI455X): state (8MB) is L2-resident; minimize full-state passes.
//   k_prep   : cos/sin/Rz-angle tables (indexed by bit position), zero accums
//   k_init   : depth-0 analytically (product state * diag phase, CNOT-perm gather)
//   k_ry_low : depth-1 Ry on bits 0..11 — three WMMA_F32_16X16X4_F32 rounds in LDS
//   k_ry_high: depth-1 Ry on bits 12..19 — two WMMA rounds + fused Rz phase
//   k_fin    : CNOT-perm gather -> d_out state + signed-prob reduction
//   k_combine: eigval = sum coefs * (e0 .* e1)

#define SUB_Q 20
#define DIMQ  (1u << SUB_Q)      // 1048576
#define TBL_OFF (2u * DIMQ)      // floats: working state occupies [0, 2*DIMQ)

typedef __attribute__((ext_vector_type(2))) float v2f;
typedef __attribute__((ext_vector_type(8))) float v8f;

// Composed CNOT chain (q=0..18, control bit 19-q, target 18-q) as a source-index map.
__device__ __forceinline__ unsigned cnot_perm(unsigned i) {
  unsigned j = i;
#pragma unroll
  for (int q = 18; q >= 0; --q)
    j ^= ((j >> (19 - q)) & 1u) << (18 - q);
  return j;
}

// Apply 16x16 real matrix U = Kron(Ry(p0+3..p0)) along LDS axis of stride 2^sshift.
// Tile of 8192 complex in LDS planes lre/lim; 512 columns; 8 waves x 4 column-tiles.
// WMMA f32 16x16x4: A lane{M=lane&15, K=2*(lane>>4)+v}; B lane{N=lane&15, K=2*(lane>>4)+v};
// D lane{N=lane&15, M=8*(lane>>4)+v}.  D = sum_kk A_chunk * B_chunk (4 chunks), re & im.
__device__ __forceinline__ void ry_round(float* lre, float* lim,
                                         const float* cs, const float* sn,
                                         int lane, int wave, int sshift) {
  const int m = lane & 15, half = lane >> 4;
  v2f a[4];
#pragma unroll
  for (int kk = 0; kk < 4; ++kk) {
    int k0 = 4 * kk + 2 * half;
    float u0 = 1.f, u1 = 1.f;
#pragma unroll
    for (int jj = 0; jj < 4; ++jj) {
      int mb = (m >> jj) & 1;
      int b0 = (k0 >> jj) & 1;
      int b1 = ((k0 + 1) >> jj) & 1;
      float c = cs[jj], s = sn[jj];
      u0 *= (mb == b0) ? c : (mb ? s : -s);   // Ry: [[c,-s],[s,c]]
      u1 *= (mb == b1) ? c : (mb ? s : -s);
    }
    a[kk][0] = u0; a[kk][1] = u1;
  }
  const int S = 1 << sshift;
#pragma unroll
  for (int t8 = 0; t8 < 4; ++t8) {
    int t = wave * 4 + t8;
    int c = t * 16 + m;                                   // this lane's column
    int colAddr = ((c >> sshift) << (sshift + 4)) + (c & (S - 1));
    v8f dre = {}; v8f dimm = {};
#pragma unroll
    for (int kk = 0; kk < 4; ++kk) {
      int k0 = 4 * kk + 2 * half;
      int ad0 = colAddr + (k0 << sshift);
      int ad1 = ad0 + S;
      v2f br, bi;
      br[0] = lre[ad0]; br[1] = lre[ad1];
      bi[0] = lim[ad0]; bi[1] = lim[ad1];
      dre  = __builtin_amdgcn_wmma_f32_16x16x4_f32(false, a[kk], false, br, (short)0, dre,  false, false);
      dimm = __builtin_amdgcn_wmma_f32_16x16x4_f32(false, a[kk], false, bi, (short)0, dimm, false, false);
    }
#pragma unroll
    for (int v = 0; v < 8; ++v) {
      int M = 8 * half + v;
      int ad = colAddr + (M << sshift);
      lre[ad] = dre[v];
      lim[ad] = dimm[v];
    }
  }
}

// Tables: cosT/sinT/bzT indexed [(part*2+d)*20 + bitpos p], p = 19 - q.
__global__ void k_prep(const float* params, float* ws) {
  float* cosT = ws + TBL_OFF;
  float* sinT = cosT + 80;
  float* bzT  = sinT + 80;
  float* accG = bzT + 80;
  int idx = threadIdx.x;
  if (idx < 80) {
    int part = idx / 40, rem = idx % 40, d = rem / 20, q = rem % 20;
    int p = 19 - q;
    float ay = params[part * 80 + d * 40 + q] * 0.5f;
    float az = params[part * 80 + d * 40 + 20 + q] * 0.5f;
    int o = (part * 2 + d) * 20 + p;
    cosT[o] = cosf(ay);
    sinT[o] = sinf(ay);
    bzT[o]  = az;
  }
  if (idx >= 80 && idx < 92) accG[idx - 80] = 0.f;
}

// Depth-0 layer analytically: out[i] = amp(j)*e^{i phi(j)}, j = cnot_perm(i).
__global__ void __launch_bounds__(256) k_init(float* state, const float* tbl, int part) {
  const float* cosT = tbl;
  const float* sinT = tbl + 80;
  const float* bzT  = tbl + 160;
  unsigned i = blockIdx.x * 256u + threadIdx.x;
  int o = (part * 2 + 0) * 20;
  unsigned j = cnot_perm(i);
  float amp = 1.f, phi = 0.f;
#pragma unroll
  for (int p = 0; p < 20; ++p) {
    int bit = (j >> p) & 1;
    amp *= bit ? sinT[o + p] : cosT[o + p];
    phi += bit ? bzT[o + p] : -bzT[o + p];
  }
  float sp, cp;
  __sincosf(phi, &sp, &cp);
  ((float2*)state)[i] = make_float2(amp * cp, amp * sp);
}

// Depth-1 Ry on bits 0..11: contiguous 8192-complex tiles, rounds at stride 1,16,256.
__global__ void __launch_bounds__(256) k_ry_low(float* state, const float* tbl, int part) {
  extern __shared__ float lds[];
  float* lre = lds;
  float* lim = lds + 8192;
  const float* cosT = tbl;
  const float* sinT = tbl + 80;
  unsigned base = blockIdx.x * 8192u;
  int tid = threadIdx.x;
  for (int e = tid; e < 8192; e += 256) {
    float2 v = ((const float2*)state)[base + e];
    lre[e] = v.x; lim[e] = v.y;
  }
  __syncthreads();
  int lane = tid & 31, wave = tid >> 5;
  int od = (part * 2 + 1) * 20;
#pragma unroll
  for (int r = 0; r < 3; ++r) {
    float cs[4], sn[4];
#pragma unroll
    for (int jj = 0; jj < 4; ++jj) { cs[jj] = cosT[od + 4 * r + jj]; sn[jj] = sinT[od + 4 * r + jj]; }
    ry_round(lre, lim, cs, sn, lane, wave, 4 * r);
    __syncthreads();
  }
  for (int e = tid; e < 8192; e += 256) {
    ((float2*)state)[base + e] = make_float2(lre[e], lim[e]);
  }
}

// Depth-1 Ry on bits 12..19 (tile: l=bits0..4 x h=bits12..19), + fused Rz phase.
__global__ void __launch_bounds__(256) k_ry_high(float* state, const float* tbl, int part) {
  extern __shared__ float lds[];
  float* lre = lds;
  float* lim = lds + 8192;
  const float* cosT = tbl;
  const float* sinT = tbl + 80;
  const float* bzT  = tbl + 160;
  int b = blockIdx.x;            // bits 5..11
  int tid = threadIdx.x;
  for (int e = 0; e < 32; ++e) {
    int f = e * 256 + tid;
    int h = f >> 5, l = f & 31;
    unsigned g = ((unsigned)h << 12) + ((unsigned)b << 5) + (unsigned)l;
    float2 v = ((const float2*)state)[g];
    lre[f] = v.x; lim[f] = v.y;
  }
  __syncthreads();
  int lane = tid & 31, wave = tid >> 5;
  int od = (part * 2 + 1) * 20;
  {
    float cs[4], sn[4];
#pragma unroll
    for (int jj = 0; jj < 4; ++jj) { cs[jj] = cosT[od + 12 + jj]; sn[jj] = sinT[od + 12 + jj]; }
    ry_round(lre, lim, cs, sn, lane, wave, 5);    // bits 12..15, LDS stride 32
    __syncthreads();
  }
  {
    float cs[4], sn[4];
#pragma unroll
    for (int jj = 0; jj < 4; ++jj) { cs[jj] = cosT[od + 16 + jj]; sn[jj] = sinT[od + 16 + jj]; }
    ry_round(lre, lim, cs, sn, lane, wave, 9);    // bits 16..19, LDS stride 512
    __syncthreads();
  }
  for (int e = 0; e < 32; ++e) {
    int f = e * 256 + tid;
    int h = f >> 5, l = f & 31;
    unsigned g = ((unsigned)h << 12) + ((unsigned)b << 5) + (unsigned)l;
    float phi = 0.f;
#pragma unroll
    for (int p = 0; p < 20; ++p)
      phi += ((g >> p) & 1u) ? bzT[od + p] : -bzT[od + p];
    float sp, cp;
    __sincosf(phi, &sp, &cp);
    float re = lre[f], im = lim[f];
    ((float2*)state)[g] = make_float2(re * cp - im * sp, re * sp + im * cp);
  }
}

// Final CNOT-chain gather -> d_out state, plus 6 signed-probability reductions.
__global__ void __launch_bounds__(256) k_fin(const float* state, float* out,
                                             const int* zmasks, float* accG, int part) {
  __shared__ float red[256];
  unsigned start = blockIdx.x * 256u + threadIdx.x;
  float acc[6] = {0.f, 0.f, 0.f, 0.f, 0.f, 0.f};
  unsigned mk[6];
#pragma unroll
  for (int t = 0; t < 6; ++t) mk[t] = (unsigned)zmasks[part * 6 + t];
  for (unsigned i = start; i < DIMQ; i += 512u * 256u) {
    unsigned j = cnot_perm(i);
    float2 v = ((const float2*)state)[j];
    size_t ob = 1 + (size_t)part * 2 * DIMQ + 2 * (size_t)i;  // out+1 is 8B-misaligned: scalar stores
    out[ob]     = v.x;
    out[ob + 1] = v.y;
    float pr = v.x * v.x + v.y * v.y;
#pragma unroll
    for (int t = 0; t < 6; ++t)
      acc[t] += (__popc(i & mk[t]) & 1) ? -pr : pr;
  }
  for (int t = 0; t < 6; ++t) {
    red[threadIdx.x] = acc[t];
    __syncthreads();
    for (int s = 128; s > 0; s >>= 1) {
      if (threadIdx.x < (unsigned)s) red[threadIdx.x] += red[threadIdx.x + s];
      __syncthreads();
    }
    if (threadIdx.x == 0) atomicAdd(&accG[part * 6 + t], red[0]);
    __syncthreads();
  }
}

__global__ void k_combine(const float* coefs, const float* accG, float* out) {
  if (threadIdx.x == 0) {
    float e = 0.f;
    for (int t = 0; t < 6; ++t) e += coefs[t] * accG[t] * accG[6 + t];
    out[0] = e;
  }
}

extern "C" void kernel_launch(void* const* d_in, const int* in_sizes, int n_in,
                              void* d_out, int out_size, void* d_ws, size_t ws_size,
                              hipStream_t stream) {
  const float* params = (const float*)d_in[0];  // 160 floats
  const float* coefs  = (const float*)d_in[1];  // 6 floats
  const int*   zmasks = (const int*)d_in[2];    // 2x6 int32
  float* ws    = (float*)d_ws;
  float* state = ws;                 // one 8MB working state, parts run sequentially
  float* tbl   = ws + TBL_OFF;       // 240 floats of tables
  float* accG  = tbl + 240;          // 12 accumulators
  float* out   = (float*)d_out;      // [eig, state0(re,im)x2^20, state1(re,im)x2^20]

  k_prep<<<1, 128, 0, stream>>>(params, ws);
  for (int part = 0; part < 2; ++part) {
    k_init   <<<DIMQ / 256,  256, 0,     stream>>>(state, tbl, part);
    k_ry_low <<<DIMQ / 8192, 256, 65536, stream>>>(state, tbl, part);
    k_ry_high<<<DIMQ / 8192, 256, 65536, stream>>>(state, tbl, part);
    k_fin    <<<512,         256, 0,     stream>>>(state, out, zmasks, accG, part);
  }
  k_combine<<<1, 64, 0, stream>>>(coefs, accG, out);
}